// TopKPool_16372415332892
// MI455X (gfx1250) — compile-verified
//
#include <hip/hip_runtime.h>
#include <hip/hip_bf16.h>

// Problem constants (fixed by the reference).
#define NB   256   // graphs
#define NN   512   // nodes per graph
#define ND   256   // channels
#define NK   256   // kept nodes per graph (top-k)

typedef __attribute__((ext_vector_type(2))) float v2f;
typedef __attribute__((ext_vector_type(8))) float v8f;

// ---------------------------------------------------------------------------
// Kernel 1: score[i] = tanh( (x[i,:] . w) * rsqrt(w.w) )  for all B*N rows.
// One wave32 per row: each lane owns 8 floats (2 x float4) of the row.
// ---------------------------------------------------------------------------
__global__ __launch_bounds__(256) void score_kernel(
    const float* __restrict__ x, const float* __restrict__ w,
    float* __restrict__ score) {
  const int tid  = blockIdx.x * blockDim.x + threadIdx.x;
  const int row  = tid >> 5;          // one wave per row
  const int lane = threadIdx.x & 31;

  const float4* x4 = (const float4*)(x + (size_t)row * ND);
  const float4* w4 = (const float4*)w;

  float4 xa = x4[lane];
  float4 xb = x4[lane + 32];
  float4 wa = w4[lane];
  float4 wb = w4[lane + 32];

  float dot = xa.x * wa.x + xa.y * wa.y + xa.z * wa.z + xa.w * wa.w
            + xb.x * wb.x + xb.y * wb.y + xb.z * wb.z + xb.w * wb.w;
  float nrm = wa.x * wa.x + wa.y * wa.y + wa.z * wa.z + wa.w * wa.w
            + wb.x * wb.x + wb.y * wb.y + wb.z * wb.z + wb.w * wb.w;

  // wave32 butterfly reduction
  #pragma unroll
  for (int off = 16; off > 0; off >>= 1) {
    dot += __shfl_xor(dot, off, 32);
    nrm += __shfl_xor(nrm, off, 32);
  }

  if (lane == 0) score[row] = tanhf(dot * rsqrtf(nrm));
}

// ---------------------------------------------------------------------------
// Kernel 2: per-graph exact top-K by rank counting, then score-gated mean
// pool of the selected rows using V_WMMA_F32_16X16X4_F32 as a GEMV engine.
// One 256-thread block (8 wave32) per graph.
// ---------------------------------------------------------------------------
__global__ __launch_bounds__(256) void pool_kernel(
    const float* __restrict__ x, const float* __restrict__ score,
    float* __restrict__ out) {
  __shared__ float    s[NN];     // raw scores for this graph
  __shared__ float    ssel[NN];  // selected scores, compacted by rank
  __shared__ unsigned soff[NN];  // selected row byte-offsets, compacted by rank

  const int g = blockIdx.x;     // graph id
  const int t = threadIdx.x;    // 0..255
  const float* __restrict__ xg = x + (size_t)g * NN * ND;
  const char*  __restrict__ xgc = (const char*)xg;

  // Stage 512 scores into LDS with the CDNA5 async copy path.
  // GLOBAL_LOAD_ASYNC_TO_LDS_B32 (GV form): vdst = LDS byte offset,
  // vaddr = 64-bit global address; the offset: immediate is added to BOTH
  // LDS and global addresses (CDNA5 ISA), so one vgpr pair
  // serves both halves. Tracked by ASYNCcnt.
  {
    unsigned     ldsa = (unsigned)(uintptr_t)&s[t];
    const float* ga   = score + g * NN + t;
    asm volatile("global_load_async_to_lds_b32 %0, %1, off"
                 :: "v"(ldsa), "v"(ga) : "memory");
    asm volatile("global_load_async_to_lds_b32 %0, %1, off offset:1024"
                 :: "v"(ldsa), "v"(ga) : "memory");
    asm volatile("s_wait_asynccnt 0x0" ::: "memory");
  }
  __syncthreads();

  // Rank of each node = # nodes strictly better (ties: lower index wins),
  // matching jax.lax.top_k ordering. Ranks are a permutation of [0,512),
  // so rank doubles as the compacted output slot. Store the row offset
  // pre-scaled to bytes so the gather loop does 32-bit address math only.
  #pragma unroll
  for (int half = 0; half < 2; ++half) {
    const int i  = t + half * 256;
    const float si = s[i];
    int rank = 0;
    for (int j = 0; j < NN; ++j) {
      const float sj = s[j];                       // LDS broadcast read
      rank += (sj > si) | ((sj == si) & (j < i));
    }
    if (rank < NK) { ssel[rank] = si; soff[rank] = (unsigned)i * (ND * 4); }
  }
  __syncthreads();

  // Gated-sum GEMV via WMMA f32 16x16x4:
  //   pooled[n] = (1/K) * sum_k ssel[k] * x[row_k, n]
  // A (16x4): all 16 M-rows carry the same 4 scores -> all D rows identical.
  //   lanes 0-15 hold K=0 (vgpr0), K=1 (vgpr1); lanes 16-31 hold K=2,3.
  // B (4x16): vgpr0 = rows K=0 (lanes 0-15) / K=2 (lanes 16-31);
  //           vgpr1 = rows K=1 / K=3. Column n = lane & 15.
  // Each wave owns channel tiles {wave, wave+8}: the second tile is the same
  // rows at +512 bytes, folded into the load's immediate offset.
  const int wave = t >> 5;
  const int lane = t & 31;
  const int hi   = lane >> 4;        // 0: K-pair {0,1}, 1: K-pair {2,3}
  const int colL = lane & 15;

  const unsigned colb = (unsigned)((wave * 16 + colL) * 4);
  v8f c0 = {};   // channels [wave*16, wave*16+16)
  v8f c1 = {};   // channels [wave*16+128, wave*16+144)

  for (int ks = 0; ks < NK; ks += 4) {
    const int i0 = ks + hi * 2;          // K slot for vgpr0 (0 or 2)
    const int i1 = i0 + 1;               // K slot for vgpr1 (1 or 3)

    const float    a0 = ssel[i0];
    const float    a1 = ssel[i1];
    const unsigned o0 = soff[i0] + colb;
    const unsigned o1 = soff[i1] + colb;

    const float b00 = *(const float*)(xgc + o0);          // tile0, K row i0
    const float b10 = *(const float*)(xgc + o1);          // tile0, K row i1
    const float b01 = *(const float*)(xgc + o0 + 512);    // tile1, K row i0
    const float b11 = *(const float*)(xgc + o1 + 512);    // tile1, K row i1

    // Unconditional clamped prefetch ~16 K-steps ahead, WGP locality.
    const int pf = (ks + 16 < NK) ? (ks + 16) : ks;
    __builtin_prefetch(xgc + soff[pf + hi * 2] + colb, 0, 3);

    v2f A  = {a0, a1};
    v2f B0 = {b00, b10};
    v2f B1 = {b01, b11};
    // 8 args: (neg_a, A, neg_b, B, c_mod, C, reuse_a, reuse_b)
    c0 = __builtin_amdgcn_wmma_f32_16x16x4_f32(false, A, false, B0,
                                               (short)0, c0, false, false);
    c1 = __builtin_amdgcn_wmma_f32_16x16x4_f32(false, A, false, B1,
                                               (short)0, c1, false, false);
  }

  // All D rows identical; row M=0 lives in c[0] on lanes 0-15.
  if (lane < 16) {
    out[g * ND + wave * 16 + lane]       = c0[0] * (1.0f / (float)NK);
    out[g * ND + wave * 16 + 128 + lane] = c1[0] * (1.0f / (float)NK);
  }
}

// ---------------------------------------------------------------------------
extern "C" void kernel_launch(void* const* d_in, const int* in_sizes, int n_in,
                              void* d_out, int out_size, void* d_ws, size_t ws_size,
                              hipStream_t stream) {
  (void)in_sizes; (void)n_in; (void)out_size; (void)ws_size;

  const float* x = (const float*)d_in[0];   // [B*N, D] float32
  const float* w = (const float*)d_in[1];   // [D]      float32
  // d_in[2] (edge_index) and d_in[3] (batch) are dead in the reference.

  float* scores = (float*)d_ws;             // B*N floats = 512 KB scratch
  float* out    = (float*)d_out;            // [B, D] float32

  // Kernel 1: one wave per row -> (B*N) waves = (B*N)*32 threads.
  const int rows    = NB * NN;              // 131072
  const int threads = 256;
  const int blocks1 = rows * 32 / threads;  // 16384
  score_kernel<<<blocks1, threads, 0, stream>>>(x, w, scores);

  // Kernel 2: one block per graph.
  pool_kernel<<<NB, threads, 0, stream>>>(x, scores, out);
}